// dirpooling_82566451298690
// MI455X (gfx1250) — compile-verified
//
#include <hip/hip_runtime.h>

// ---------------------------------------------------------------------------
// dirpooling: out = relu(repeat(conv3x3(sum_c x, 6 fixed stencils), 32))
// x: (8,64,256,256) f32   out: (8,192,256,256) f32   ws: xs = (8,256,256) f32
// Pure bandwidth problem: 134MB read + 403MB NT-store write, ~23us floor.
// ---------------------------------------------------------------------------

typedef __attribute__((ext_vector_type(4))) float f4;
typedef __attribute__((ext_vector_type(2))) float v2f;
typedef __attribute__((ext_vector_type(8))) float v8f;
typedef __attribute__((ext_vector_type(4))) int   v4i;

// async-to-LDS builtin parameter types (per hipcc diagnostic: int4 in as(1)/as(3))
typedef __attribute__((address_space(1))) v4i* g128_t;
typedef __attribute__((address_space(3))) v4i* l128_t;

#define NB 8
#define NC 64
#define OC 192
#define PLANE 65536            // 256*256
#define XSTR 264               // LDS row stride (floats): 16B-aligned payload at +4
#define TROWS 4                // output rows per block in pass 2

#if __has_builtin(__builtin_amdgcn_global_load_async_to_lds_b128)
#define HAVE_ASYNC 1
#else
#define HAVE_ASYNC 0
#endif

#if __has_builtin(__builtin_amdgcn_wmma_f32_16x16x4_f32)
#define HAVE_WMMA 1
#else
#define HAVE_WMMA 0
#endif

// per-direction tap masks over the 9 taps t = (dy+1)*3 + (dx+1)
__device__ const unsigned g_dirmask[6] = {
    0x111u, // main diagonal: t0,t4,t8
    0x054u, // anti diagonal: t2,t4,t6
    0x155u, // X:             t0,t2,t4,t6,t8
    0x0BAu, // +:             t1,t3,t4,t5,t7
    0x038u, // horizontal:    t3,t4,t5
    0x092u  // vertical:      t1,t4,t7
};

__device__ __forceinline__ float dirw(int m, int t) {
    return (m < 6 && t < 9) ? (float)((g_dirmask[m] >> t) & 1u) : 0.0f;
}

// ---------------- Pass 1: channel sum (134MB NT read -> 2MB write) ----------
__global__ __launch_bounds__(256) void k_chansum(const float* __restrict__ x,
                                                 float* __restrict__ xs) {
    int t = blockIdx.x * 256 + threadIdx.x;   // 0 .. NB*PLANE/4 - 1
    int b = t >> 14;                          // PLANE/4 == 16384
    int g = t & 16383;
    const f4* px = (const f4*)x + (size_t)b * NC * (PLANE / 4) + g;
    f4 acc = {0.f, 0.f, 0.f, 0.f};
#pragma unroll 16
    for (int c = 0; c < NC; ++c) {
        f4 v = __builtin_nontemporal_load(px + (size_t)c * (PLANE / 4));
        acc += v;                              // stride-256KB streams, coalesced per wave
    }
    ((f4*)xs)[t] = acc;                        // regular store: keep xs hot in L2
}

// -------- Pass 2: stencil (async LDS stage + WMMA) + 32x replicate NT store --
__global__ __launch_bounds__(256) void k_stencil_rep(const float* __restrict__ xs,
                                                     float* __restrict__ out) {
    __shared__ float tile[6 * XSTR];          // 6 xs rows, zero halo at cols 3 / 260
    __shared__ float dirs[6 * TROWS * 256];   // 6 relu'd directional maps, 24KB

    const int t  = threadIdx.x;
    const int b  = blockIdx.y;
    const int y0 = blockIdx.x * TROWS;

    // zero the two halo columns of each row
    if (t < 12) tile[(t >> 1) * XSTR + ((t & 1) ? 260 : 3)] = 0.0f;
    // zero out-of-range boundary rows (only r=0 / r=5 can be OOB)
    if (y0 == 0)             tile[0 * XSTR + 4 + t] = 0.0f;
    if (y0 + TROWS >= 256)   tile[5 * XSTR + 4 + t] = 0.0f;

    // stage 6 rows x 256 floats (384 b128 slots) into LDS
    const float* xsb = xs + ((size_t)b << 16);
    for (int s = t; s < 384; s += 256) {
        int r = s >> 6, c4 = s & 63;
        int gy = y0 - 1 + r;
        if (gy >= 0 && gy <= 255) {
            const f4* src = (const f4*)(xsb + (gy << 8)) + c4;
            float* dst = &tile[r * XSTR + 4 + c4 * 4];       // 16B aligned
#if HAVE_ASYNC
            __builtin_amdgcn_global_load_async_to_lds_b128((g128_t)src, (l128_t)dst,
                                                           0, 0);
#else
            *(f4*)dst = *src;
#endif
        }
    }
#if HAVE_ASYNC
#if __has_builtin(__builtin_amdgcn_s_wait_asynccnt)
    __builtin_amdgcn_s_wait_asynccnt(0);
#else
    asm volatile("s_wait_asynccnt 0" ::: "memory");
#endif
#endif
    __syncthreads();

#if HAVE_WMMA
    // Stencil as GEMM: D(16 dirs x 16 px) = A(16x12 weights) * B(12 taps x 16 px)
    // via 3 chained v_wmma_f32_16x16x4_f32. Rows M=0..5 of D are the 6 dirs.
    {
        const int lane = t & 31;
        const int wv   = t >> 5;        // 8 waves
        const int n    = lane & 15;     // pixel within group / weight row M
        const int hi   = lane >> 4;     // K-half select (K=0,1 vs K=2,3)
        // A is constant per lane across all groups: hoist the 6 weight values
        float a0[3], a1[3];
#pragma unroll
        for (int s = 0; s < 3; ++s) {
            int k0 = 4 * s + 2 * hi;
            a0[s] = dirw(n, k0);
            a1[s] = dirw(n, k0 + 1);
        }
        for (int it = 0; it < 8; ++it) {
            const int g  = wv * 8 + it;         // 64 groups of 16 pixels
            const int lr = g >> 4;              // tile row 0..3
            const int xb = (g & 15) << 4;       // pixel column base
            v8f acc = {0.f, 0.f, 0.f, 0.f, 0.f, 0.f, 0.f, 0.f};
#pragma unroll
            for (int s = 0; s < 3; ++s) {
                const int k0 = 4 * s + 2 * hi;  // tap indices for this lane half
                v2f A, B;
                A.x = a0[s];
                A.y = a1[s];
                const int t0 = k0, t1 = k0 + 1;
                B.x = (t0 < 9) ? tile[(lr + t0 / 3) * XSTR + 3 + xb + n + t0 % 3] : 0.0f;
                B.y = (t1 < 9) ? tile[(lr + t1 / 3) * XSTR + 3 + xb + n + t1 % 3] : 0.0f;
                acc = __builtin_amdgcn_wmma_f32_16x16x4_f32(
                    false, A, false, B, (short)0, acc, false, false);
            }
            if (lane < 16) {   // lanes 0-15 hold M=0..7 in acc[0..7]
#pragma unroll
                for (int m = 0; m < 6; ++m)
                    dirs[m * 1024 + lr * 256 + xb + n] = fmaxf(acc[m], 0.0f);
            }
        }
    }
#else
    // VALU fallback stencil
    for (int i = 0; i < 4; ++i) {
        int p   = i * 256 + t;
        int lr  = p >> 8, col = p & 255;
        int cr  = (lr + 1) * XSTR + 4 + col;
        float uc = tile[cr - XSTR], dc = tile[cr + XSTR];
        float ul = tile[cr - XSTR - 1], ur = tile[cr - XSTR + 1];
        float dl = tile[cr + XSTR - 1], dr = tile[cr + XSTR + 1];
        float lf = tile[cr - 1], rt = tile[cr + 1], cc = tile[cr];
        dirs[0 * 1024 + p] = fmaxf(ul + cc + dr, 0.f);
        dirs[1 * 1024 + p] = fmaxf(ur + cc + dl, 0.f);
        dirs[2 * 1024 + p] = fmaxf(ul + ur + cc + dl + dr, 0.f);
        dirs[3 * 1024 + p] = fmaxf(uc + lf + cc + rt + dc, 0.f);
        dirs[4 * 1024 + p] = fmaxf(lf + cc + rt, 0.f);
        dirs[5 * 1024 + p] = fmaxf(uc + cc + dc, 0.f);
    }
#endif
    __syncthreads();

    // Replicate each dir map 32x with streaming NT b128 stores (403MB total).
    // Inner 32 stores span 31*256KB < 8MB -> fits the signed 24-bit imm offset.
    float* obase = out + (((size_t)b * OC) << 16) + (y0 << 8) + t * 4;
#pragma unroll
    for (int d = 0; d < 6; ++d) {
        f4 v = *(const f4*)&dirs[d * 1024 + t * 4];
        float* p0 = obase + (((size_t)d * 32) << 16);
#pragma unroll
        for (int r = 0; r < 32; ++r)
            __builtin_nontemporal_store(v, (f4*)(p0 + ((size_t)r << 16)));
    }
}

// ---------------------------------------------------------------------------
extern "C" void kernel_launch(void* const* d_in, const int* in_sizes, int n_in,
                              void* d_out, int out_size, void* d_ws, size_t ws_size,
                              hipStream_t stream) {
    (void)in_sizes; (void)n_in; (void)out_size; (void)ws_size;
    const float* x  = (const float*)d_in[0];
    float*       out = (float*)d_out;
    float*       xs  = (float*)d_ws;   // needs 8*256*256*4 = 2MB of scratch

    k_chansum<<<dim3((NB * PLANE / 4) / 256), 256, 0, stream>>>(x, xs);
    k_stencil_rep<<<dim3(256 / TROWS, NB), 256, 0, stream>>>(xs, out);
}